// MultiGCN_20040317403628
// MI455X (gfx1250) — compile-verified
//
#include <hip/hip_runtime.h>
#include <hip/hip_bf16.h>

#define N_NODES 100000
#define FEAT 128

typedef __attribute__((ext_vector_type(2))) float v2f;
typedef __attribute__((ext_vector_type(8))) float v8f;

// ---------------- degree / normalization ----------------

__global__ void zero_u32_kernel(unsigned* __restrict__ p, int n) {
  int i = blockIdx.x * blockDim.x + threadIdx.x;
  if (i < n) p[i] = 0u;
}

__global__ void degree_kernel(const int* __restrict__ cols,
                              unsigned* __restrict__ deg, int E) {
  int e = blockIdx.x * blockDim.x + threadIdx.x;
  if (e < E) atomicAdd(&deg[cols[e]], 1u);
}

__global__ void make_dis_kernel(float* __restrict__ dis, int n) {
  int i = blockIdx.x * blockDim.x + threadIdx.x;
  if (i < n) {
    unsigned d = ((const unsigned*)dis)[i];
    dis[i] = rsqrtf((float)(d + 1u));   // +1 self loop => deg >= 1 always
  }
}

// ---------------- fp32 WMMA GEMM: H[M,128] = act(X)[M,128] @ W[128,128] ----------------
// One wave -> 16x64 output strip (four 16x16 N-tiles); K-loop of 32 x V_WMMA_F32_16X16X4_F32
// per tile. W is staged in LDS in two 64-row chunks, packed as (k,k+1) float2 pairs with
// pair-row stride 144 (2*144 = 288 ≡ 16 mod 32 pair-banks) so each ds_load_b64 is
// bank-conflict free across the lane halves.
//
// A layout: lane l holds row m0+(l&15); VGPR v -> K = v + 2*(l>>4).
// B layout (symmetric): lane l holds col n;  VGPR v -> K = v + 2*(l>>4).
// C/D layout: VGPR r -> M = r + 8*(l>>4), N = l&15.

#define WPAIR_STRIDE 144   // float2 elements per packed pair-row (128 + 16 pad)
#define KCHUNK 64          // K rows staged per LDS pass (32 pair-rows)

template <bool RELU>
__global__ void gcn_gemm_wmma(const float* __restrict__ X,
                              const float* __restrict__ W,
                              float* __restrict__ H) {
  __shared__ v2f wt[(KCHUNK / 2) * WPAIR_STRIDE];   // 32 * 144 * 8B = 36 KB

  const int lane   = threadIdx.x & 31;
  const int lane15 = lane & 15;
  const int laneHi = lane >> 4;                      // 0 or 1
  const int kOff   = laneHi << 1;                    // 0 or 2
  const int wave   = blockIdx.x * (blockDim.x >> 5) + (threadIdx.x >> 5);
  const int mTile  = wave >> 1;                      // 16-row tile (exact: N%16==0)
  const int nHalf  = wave & 1;                       // which 64-column half
  const int n0     = nHalf << 6;

  const int mRow = mTile * 16 + lane15;
  const float* __restrict__ xrow = X + (size_t)mRow * FEAT;

  v8f acc0 = {}, acc1 = {}, acc2 = {}, acc3 = {};

  for (int kb = 0; kb < FEAT; kb += KCHUNK) {
    // ---- cooperative stage of W[kb:kb+64, :] into packed-pair LDS ----
    for (int idx = threadIdx.x; idx < (KCHUNK / 2) * FEAT; idx += blockDim.x) {
      int kp = idx >> 7;          // local pair-row 0..31
      int n  = idx & 127;
      int gk = kb + (kp << 1);
      v2f p;
      p.x = W[(size_t)gk * FEAT + n];
      p.y = W[(size_t)(gk + 1) * FEAT + n];
      wt[kp * WPAIR_STRIDE + n] = p;
    }
    __syncthreads();

    if (kb + KCHUNK < FEAT)
      __builtin_prefetch(xrow + kb + KCHUNK, 0, 0);  // global_prefetch_b8 for next A chunk

#pragma unroll 4
    for (int k = 0; k < KCHUNK; k += 4) {
      v2f a;
      a.x = xrow[kb + k + kOff];
      a.y = xrow[kb + k + kOff + 1];
      if (RELU) { a.x = fmaxf(a.x, 0.0f); a.y = fmaxf(a.y, 0.0f); }

      const v2f* bp = wt + ((k >> 1) + laneHi) * WPAIR_STRIDE + n0 + lane15;
      acc0 = __builtin_amdgcn_wmma_f32_16x16x4_f32(false, a, false, bp[0],
                                                   (short)0, acc0, false, false);
      acc1 = __builtin_amdgcn_wmma_f32_16x16x4_f32(false, a, false, bp[16],
                                                   (short)0, acc1, false, false);
      acc2 = __builtin_amdgcn_wmma_f32_16x16x4_f32(false, a, false, bp[32],
                                                   (short)0, acc2, false, false);
      acc3 = __builtin_amdgcn_wmma_f32_16x16x4_f32(false, a, false, bp[48],
                                                   (short)0, acc3, false, false);
    }
    __syncthreads();
  }

  const int rBase = mTile * 16 + (laneHi << 3);
  float* __restrict__ hOut = H + (size_t)rBase * FEAT + n0 + lane15;
#pragma unroll
  for (int r = 0; r < 8; ++r) {
    float* p = hOut + (size_t)r * FEAT;
    p[0]  = acc0[r];
    p[16] = acc1[r];
    p[32] = acc2[r];
    p[48] = acc3[r];
  }
}

// ---------------- self-loop + bias init: out[i][:] = b + dis[i]^2 * h[i][:] ----------------

__global__ void selfloop_bias_kernel(const float* __restrict__ H,
                                     const float* __restrict__ dis,
                                     const float* __restrict__ bias,
                                     float* __restrict__ out) {
  int idx = blockIdx.x * blockDim.x + threadIdx.x;   // one float4 (4 features)
  const int total = N_NODES * (FEAT / 4);
  if (idx >= total) return;
  int node = idx >> 5;       // 32 float4 per node
  int f4   = idx & 31;
  float d = dis[node];
  float w = d * d;
  float4 v = ((const float4*)H)[idx];
  float4 b = ((const float4*)bias)[f4];
  float4 o;
  o.x = fmaf(w, v.x, b.x);
  o.y = fmaf(w, v.y, b.y);
  o.z = fmaf(w, v.z, b.z);
  o.w = fmaf(w, v.w, b.w);
  ((float4*)out)[idx] = o;
}

// ---------------- edge propagate: out[col] += dis[row]*dis[col] * h[row] ----------------
// One wave per edge: 32 lanes x float4 = 128 features; hardware fp32 atomics into L2.

__device__ __forceinline__ void atomicAddF32(float* p, float v) {
  __hip_atomic_fetch_add(p, v, __ATOMIC_RELAXED, __HIP_MEMORY_SCOPE_AGENT);
}

__global__ void propagate_kernel(const int* __restrict__ rows,
                                 const int* __restrict__ cols,
                                 const float* __restrict__ dis,
                                 const float* __restrict__ H,
                                 float* __restrict__ out, int E) {
  const int lane = threadIdx.x & 31;
  const int e = blockIdx.x * (blockDim.x >> 5) + (threadIdx.x >> 5);
  if (e >= E) return;
  const int r = rows[e];
  const int c = cols[e];
  const float w = dis[r] * dis[c];
  float4 v = ((const float4*)(H + (size_t)r * FEAT))[lane];
  float* dst = out + (size_t)c * FEAT + (lane << 2);
  atomicAddF32(dst + 0, w * v.x);
  atomicAddF32(dst + 1, w * v.y);
  atomicAddF32(dst + 2, w * v.z);
  atomicAddF32(dst + 3, w * v.w);
}

// ---------------- launch ----------------

extern "C" void kernel_launch(void* const* d_in, const int* in_sizes, int n_in,
                              void* d_out, int out_size, void* d_ws, size_t ws_size,
                              hipStream_t stream) {
  (void)n_in; (void)out_size; (void)ws_size;
  const float* x  = (const float*)d_in[0];
  const int*   ed = (const int*)d_in[1];     // [2, E] int32
  const float* W1 = (const float*)d_in[2];
  const float* b1 = (const float*)d_in[3];
  const float* W2 = (const float*)d_in[4];
  const float* b2 = (const float*)d_in[5];
  const int E = in_sizes[1] / 2;
  const int* rows = ed;        // edges[0] = sources
  const int* cols = ed + E;    // edges[1] = targets

  // workspace layout: dis (400KB) | h (51.2MB) | agg (51.2MB)
  float* dis = (float*)d_ws;
  float* h   = (float*)((char*)d_ws + (1 << 20));
  float* agg = h + (size_t)N_NODES * FEAT;
  float* out = (float*)d_out;

  const int T = 256;
  const int nBlk    = (N_NODES + T - 1) / T;
  const int eBlk    = (E + T - 1) / T;
  const int mTiles  = N_NODES / 16;            // 6250 (exact)
  const int gWaves  = mTiles * 2;              // 2 N-halves of 64 cols
  const int gemmT   = 128;                     // 4 waves/block
  const int gemmBlk = gWaves / 4;              // 3125 (exact) -> no barrier divergence
  const int initBlk = (N_NODES * (FEAT / 4) + T - 1) / T;
  const int propBlk = (E + 7) / 8;             // 8 edges (waves) per block

  // normalization: deg over targets (+1 self loop) -> rsqrt
  zero_u32_kernel<<<nBlk, T, 0, stream>>>((unsigned*)dis, N_NODES);
  degree_kernel<<<eBlk, T, 0, stream>>>(cols, (unsigned*)dis, E);
  make_dis_kernel<<<nBlk, T, 0, stream>>>(dis, N_NODES);

  // layer 1: h = x @ W1 ; agg = b1 + dis^2*h (self loops) ; scatter-add edges
  gcn_gemm_wmma<false><<<gemmBlk, gemmT, 0, stream>>>(x, W1, h);
  selfloop_bias_kernel<<<initBlk, T, 0, stream>>>(h, dis, b1, agg);
  propagate_kernel<<<propBlk, T, 0, stream>>>(rows, cols, dis, h, agg, E);

  // layer 2: h = relu(agg) @ W2 (ReLU fused into A-load) ; out = b2 + dis^2*h ; scatter
  gcn_gemm_wmma<true><<<gemmBlk, gemmT, 0, stream>>>(agg, W2, h);
  selfloop_bias_kernel<<<initBlk, T, 0, stream>>>(h, dis, b2, out);
  propagate_kernel<<<propBlk, T, 0, stream>>>(rows, cols, dis, h, out, E);
}